// TorchLMHeadORPO_8701603741901
// MI455X (gfx1250) — compile-verified
//
#include <hip/hip_runtime.h>
#include <hip/hip_bf16.h>
#include <math.h>

// ---------------- problem constants ----------------
#define B_ROWS   4096
#define H_DIM    2048
#define V_DIM    32000
#define HALF_B   2048
#define IGNORE_INDEX (-100)
#define BETA     0.1f

// ---------------- tiling ----------------
#define BM   128           // rows per block (halves W streaming traffic vs 64)
#define BN   128           // vocab cols per block
#define KC   32            // K chunk staged in LDS
#define NVB  (V_DIM / BN)  // 250 vocab blocks
#define LDSK (KC + 4)      // 36 floats = 144B stride, 16B aligned

typedef __attribute__((ext_vector_type(2))) float v2f;
typedef __attribute__((ext_vector_type(8))) float v8f;

// =====================================================================
// Kernel 0: zero the target-logit slots (rows with IGNORE_INDEX never
// get written by kernel 1).
// =====================================================================
__global__ void orpo_init(float* tlog) {
    int i = blockIdx.x * blockDim.x + threadIdx.x;
    if (i < HALF_B) tlog[i] = 0.0f;
}

// =====================================================================
// Kernel 1: fused tile-GEMM (f32 WMMA) + per-row partial softmax stats.
// grid = (NVB, B/BM); block = 256 threads (8 waves).
// Wave w owns rows [w*16, w*16+16) x all 128 cols -> 8 accumulators.
// =====================================================================
__global__ __launch_bounds__(256)
void orpo_gemm_stats(const float* __restrict__ x,
                     const float* __restrict__ W,
                     const long long* __restrict__ y,
                     float* __restrict__ pmax,
                     float* __restrict__ psum,
                     float* __restrict__ plog,
                     float* __restrict__ tlog) {
    __shared__ float As[BM][LDSK];      // x tile [row][k]
    __shared__ float Ws[BN][LDSK];      // W tile [vcol][k]

    const int vb   = blockIdx.x;        // vocab block
    const int rb   = blockIdx.y;        // row block
    const int tid  = threadIdx.x;
    const int w    = tid >> 5;          // wave 0..7 -> row subtile
    const int lane = tid & 31;
    const int hg   = lane >> 4;         // half-group 0/1
    const int l16  = lane & 15;

    v8f acc[8];
    #pragma unroll
    for (int t = 0; t < 8; ++t) acc[t] = (v8f){0,0,0,0,0,0,0,0};

    // Per-thread staging coordinates: 128 rows x 8 float4 per row
    // = 1024 float4 -> 4 per thread per tile.
    for (int k0 = 0; k0 < H_DIM; k0 += KC) {
        #pragma unroll
        for (int it = 0; it < 4; ++it) {
            const int i   = tid + it * 256;
            const int row = i >> 3;
            const int c4  = (i & 7) << 2;
            const float4 va = *(const float4*)(x + (size_t)(rb * BM + row) * H_DIM + k0 + c4);
            *(float4*)(&As[row][c4]) = va;
            const float4 vw = *(const float4*)(W + (size_t)(vb * BN + row) * H_DIM + k0 + c4);
            *(float4*)(&Ws[row][c4]) = vw;
            // pull next K-chunk toward the caches while we compute
            if (k0 + KC < H_DIM) {
                __builtin_prefetch(x + (size_t)(rb * BM + row) * H_DIM + k0 + KC + c4, 0, 3);
                __builtin_prefetch(W + (size_t)(vb * BN + row) * H_DIM + k0 + KC + c4, 0, 3);
            }
        }
        __syncthreads();

        // ---- compute: K in steps of 4 (V_WMMA_F32_16X16X4_F32) ----
        #pragma unroll 2
        for (int k4 = 0; k4 < KC; k4 += 4) {
            const int kb = k4 + (hg << 1);
            // A fragment: row m = w*16 + l16, K = kb, kb+1 (lane-half -> K pair)
            v2f a;
            a.x = As[w * 16 + l16][kb];
            a.y = As[w * 16 + l16][kb + 1];
            #pragma unroll
            for (int t = 0; t < 8; ++t) {
                const int nl = t * 16 + l16;            // local vocab col
                v2f b;
                b.x = Ws[nl][kb];
                b.y = Ws[nl][kb + 1];
                acc[t] = __builtin_amdgcn_wmma_f32_16x16x4_f32(
                    false, a, false, b, (short)0, acc[t], false, false);
            }
        }
        __syncthreads();
    }

    // -------- per-row reduction over all 128 columns (intra-wave) ----
    // C layout: VGPR i, lanes 0-15 -> M=i (N=lane), lanes 16-31 -> M=i+8.
    #pragma unroll
    for (int i = 0; i < 8; ++i) {
        float mx = acc[0][i];
        #pragma unroll
        for (int t = 1; t < 8; ++t) mx = fmaxf(mx, acc[t][i]);
        #pragma unroll
        for (int m = 1; m < 16; m <<= 1)
            mx = fmaxf(mx, __shfl_xor(mx, m, 32));

        float se = 0.0f, sl = 0.0f;
        #pragma unroll
        for (int t = 0; t < 8; ++t) {
            se += __expf(acc[t][i] - mx);
            sl += acc[t][i];
        }
        #pragma unroll
        for (int m = 1; m < 16; m <<= 1) {
            se += __shfl_xor(se, m, 32);
            sl += __shfl_xor(sl, m, 32);
        }

        const int lr   = w * 16 + i + (hg << 3);        // local row 0..127
        const int grow = rb * BM + lr;
        if (l16 == 0) {
            const size_t idx = (size_t)grow * NVB + vb;
            pmax[idx] = mx;
            psum[idx] = se;
            plog[idx] = sl;
        }
        // target-logit scatter (first half only)
        if (grow < HALF_B) {
            const long long ty = y[grow];
            #pragma unroll
            for (int t = 0; t < 8; ++t) {
                const int gv = vb * BN + t * 16 + l16;
                if (ty == (long long)gv) tlog[grow] = acc[t][i];
            }
        }
    }
}

// =====================================================================
// Kernel 2: per-row merge of NVB partials -> lse, mean logprob.
// One wave per row; block = 256 threads = 8 rows.
// =====================================================================
__global__ __launch_bounds__(256)
void orpo_row_reduce(const float* __restrict__ pmax,
                     const float* __restrict__ psum,
                     const float* __restrict__ plog,
                     float* __restrict__ lse,
                     float* __restrict__ mlp) {
    const int lane = threadIdx.x & 31;
    const int row  = blockIdx.x * 8 + (threadIdx.x >> 5);

    float m = -INFINITY, s = 0.0f, l = 0.0f;
    const size_t base = (size_t)row * NVB;
    for (int vb = lane; vb < NVB; vb += 32) {
        const float pm = pmax[base + vb];
        const float ps = psum[base + vb];
        const float nm = fmaxf(m, pm);
        s = s * __expf(m - nm) + ps * __expf(pm - nm);
        m = nm;
        l += plog[base + vb];
    }
    #pragma unroll
    for (int msk = 1; msk < 32; msk <<= 1) {
        const float om = __shfl_xor(m, msk, 32);
        const float os = __shfl_xor(s, msk, 32);
        const float ol = __shfl_xor(l, msk, 32);
        const float nm = fmaxf(m, om);
        s = s * __expf(m - nm) + os * __expf(om - nm);
        m = nm;
        l += ol;
    }
    if (lane == 0) {
        const float L = m + __logf(s);              // logsumexp
        lse[row] = L;
        mlp[row] = l / (float)V_DIM - L;            // mean log-softmax
    }
}

// =====================================================================
// Kernel 3: final scalar. One block.
// =====================================================================
__device__ __forceinline__ float log_sigmoid(float z) {
    return (z >= 0.0f) ? -log1pf(__expf(-z)) : (z - log1pf(__expf(z)));
}

__global__ __launch_bounds__(512)
void orpo_final(const float* __restrict__ lse,
                const float* __restrict__ mlp,
                const float* __restrict__ tlog,
                const long long* __restrict__ y,
                float* __restrict__ out) {
    __shared__ float rN[512], rC[512], rO[512];
    const int t = threadIdx.x;
    float nll = 0.0f, cnt = 0.0f, ors = 0.0f;
    for (int row = t; row < HALF_B; row += 512) {
        const bool valid = (y[row] != (long long)IGNORE_INDEX);
        if (valid) {
            nll += lse[row] - tlog[row];
            cnt += 1.0f;
        }
        const float ch = mlp[row];
        const float rj = mlp[row + HALF_B];
        const float lo = (ch - rj)
                       - (log1pf(-__expf(ch)) - log1pf(-__expf(rj)));
        ors += BETA * log_sigmoid(lo);
    }
    rN[t] = nll; rC[t] = cnt; rO[t] = ors;
    __syncthreads();
    for (int st = 256; st > 0; st >>= 1) {
        if (t < st) { rN[t] += rN[t + st]; rC[t] += rC[t + st]; rO[t] += rO[t + st]; }
        __syncthreads();
    }
    if (t == 0) {
        const float ce = rN[0] / fmaxf(rC[0], 1.0f);
        out[0] = ce - rO[0] / (float)HALF_B;
    }
}

// =====================================================================
extern "C" void kernel_launch(void* const* d_in, const int* in_sizes, int n_in,
                              void* d_out, int out_size, void* d_ws, size_t ws_size,
                              hipStream_t stream) {
    const float*     x = (const float*)d_in[0];
    const float*     W = (const float*)d_in[1];
    const long long* y = (const long long*)d_in[2];
    float* out = (float*)d_out;

    float* pmax = (float*)d_ws;
    float* psum = pmax + (size_t)B_ROWS * NVB;
    float* plog = psum + (size_t)B_ROWS * NVB;
    float* tlog = plog + (size_t)B_ROWS * NVB;
    float* lse  = tlog + HALF_B;
    float* mlp  = lse  + B_ROWS;

    orpo_init<<<(HALF_B + 255) / 256, 256, 0, stream>>>(tlog);

    dim3 g1(NVB, B_ROWS / BM);
    orpo_gemm_stats<<<g1, 256, 0, stream>>>(x, W, y, pmax, psum, plog, tlog);

    orpo_row_reduce<<<B_ROWS / 8, 256, 0, stream>>>(pmax, psum, plog, lse, mlp);

    orpo_final<<<1, 512, 0, stream>>>(lse, mlp, tlog, y, out);
}